// StereoNet_8684423873310
// MI455X (gfx1250) — compile-verified
//
#include <hip/hip_runtime.h>
#include <hip/hip_bf16.h>
#include <math.h>

#define NUM_CLASS 3
#define SCORE_THRESH 0.1f
#define NMS_THRESH 0.25f
#define PRE 512
#define POST 100
#define NBINS 4096

typedef float v2f __attribute__((ext_vector_type(2)));
typedef float v8f __attribute__((ext_vector_type(8)));

__device__ __forceinline__ float sigmoidf_(float x) { return 1.0f / (1.0f + __expf(-x)); }

// Exact rotated-rect BEV intersection area (same construction as the reference:
// inside corners + edge intersections, angle-sorted shoelace). noinline: this is
// the rare slow path (~2% of pairs after the WMMA distance prefilter); keep one
// copy instead of 8x unrolled duplicates.
__device__ __attribute__((noinline)) float bev_inter_area(const float* A, const float* B) {
    float cax[4], cay[4], cbx[4], cby[4];
    const float SX[4] = {0.5f, -0.5f, -0.5f, 0.5f};
    const float SY[4] = {0.5f, 0.5f, -0.5f, -0.5f};
    float ca = cosf(A[6]), sa = sinf(A[6]);
    float cb = cosf(B[6]), sb = sinf(B[6]);
#pragma unroll
    for (int i = 0; i < 4; i++) {
        float lx = SX[i] * A[3], ly = SY[i] * A[4];
        cax[i] = A[0] + lx * ca - ly * sa;
        cay[i] = A[1] + lx * sa + ly * ca;
        lx = SX[i] * B[3]; ly = SY[i] * B[4];
        cbx[i] = B[0] + lx * cb - ly * sb;
        cby[i] = B[1] + lx * sb + ly * cb;
    }
    float px[24], py[24];
    int cnt = 0;
    // corners of A inside B
#pragma unroll
    for (int i = 0; i < 4; i++) {
        float dx = cax[i] - B[0], dy = cay[i] - B[1];
        float u = dx * cb + dy * sb, v = -dx * sb + dy * cb;
        if (fabsf(u) <= B[3] * 0.5f + 1e-5f && fabsf(v) <= B[4] * 0.5f + 1e-5f) {
            px[cnt] = cax[i]; py[cnt] = cay[i]; cnt++;
        }
    }
    // corners of B inside A
#pragma unroll
    for (int i = 0; i < 4; i++) {
        float dx = cbx[i] - A[0], dy = cby[i] - A[1];
        float u = dx * ca + dy * sa, v = -dx * sa + dy * ca;
        if (fabsf(u) <= A[3] * 0.5f + 1e-5f && fabsf(v) <= A[4] * 0.5f + 1e-5f) {
            px[cnt] = cbx[i]; py[cnt] = cby[i]; cnt++;
        }
    }
    // edge-edge intersections
#pragma unroll
    for (int i = 0; i < 4; i++) {
        float rax = cax[(i + 1) & 3] - cax[i], ray = cay[(i + 1) & 3] - cay[i];
#pragma unroll
        for (int j = 0; j < 4; j++) {
            float rbx = cbx[(j + 1) & 3] - cbx[j], rby = cby[(j + 1) & 3] - cby[j];
            float d = rax * rby - ray * rbx;
            if (fabsf(d) > 1e-8f) {
                float qx = cbx[j] - cax[i], qy = cby[j] - cay[i];
                float t = (qx * rby - qy * rbx) / d;
                float u = (qx * ray - qy * rax) / d;
                if (t >= 0.0f && t <= 1.0f && u >= 0.0f && u <= 1.0f) {
                    px[cnt] = cax[i] + t * rax; py[cnt] = cay[i] + t * ray; cnt++;
                }
            }
        }
    }
    if (cnt < 3) return 0.0f;
    float cx = 0.0f, cy = 0.0f;
    for (int i = 0; i < cnt; i++) { cx += px[i]; cy += py[i]; }
    cx /= (float)cnt; cy /= (float)cnt;
    float ang[24];
    for (int i = 0; i < cnt; i++) ang[i] = atan2f(py[i] - cy, px[i] - cx);
    for (int i = 1; i < cnt; i++) {  // insertion sort by angle (deterministic)
        float a = ang[i], x = px[i], y = py[i];
        int j = i - 1;
        while (j >= 0 && ang[j] > a) { ang[j+1]=ang[j]; px[j+1]=px[j]; py[j+1]=py[j]; j--; }
        ang[j+1] = a; px[j+1] = x; py[j+1] = y;
    }
    float area = 0.0f;
    for (int i = 0; i < cnt; i++) {
        int n = (i + 1 == cnt) ? 0 : i + 1;
        area += px[i] * py[n] - py[i] * px[n];
    }
    return 0.5f * fabsf(area);
}

__device__ __attribute__((noinline)) float iou3d_pair(const float* a, const float* b) {
    float inter = bev_inter_area(a, b);
    float zt = fminf(a[2] + a[5] * 0.5f, b[2] + b[5] * 0.5f);
    float zb = fmaxf(a[2] - a[5] * 0.5f, b[2] - b[5] * 0.5f);
    float i3 = inter * fmaxf(zt - zb, 0.0f);
    float va = a[3] * a[4] * a[5], vb = b[3] * b[4] * b[5];
    return i3 / fmaxf(va + vb - i3, 1e-8f);
}

// ---------------------------------------------------------------------------
// Kernel 1: per (batch,class) slot — sigmoid scoring, deterministic histogram-
// based top-512 selection, in-LDS bitonic sort (desc, index tiebreak), box gather.
// One 1024-thread workgroup (32 wave32) per slot.
// ---------------------------------------------------------------------------
__global__ __launch_bounds__(1024) void topk_kernel(
    const float* __restrict__ cls, const float* __restrict__ box,
    float* __restrict__ wsScore, float* __restrict__ wsBox, int N)
{
    __shared__ unsigned int hist[NBINS];
    __shared__ unsigned int wsum[33];
    __shared__ float sSc[PRE];
    __shared__ int sIx[PRE];
    __shared__ int sCut, sNeedEq, sRunA, sRunE;

    const int tid = threadIdx.x;
    const int slot = blockIdx.x;
    const int b = slot / NUM_CLASS, kcls = slot % NUM_CLASS;
    const float* clsB = cls + (size_t)b * N * NUM_CLASS;

    for (int i = tid; i < NBINS; i += 1024) hist[i] = 0;
    if (tid < PRE) { sSc[tid] = -1.0f; sIx[tid] = -1; }
    __syncthreads();

    // Pass 1: score histogram (only entries >= threshold).
    for (int i = tid; i < N; i += 1024) {
        float sc = sigmoidf_(clsB[(size_t)i * NUM_CLASS + kcls]);
        if (sc >= SCORE_THRESH) {
            int bin = (int)(sc * (float)NBINS);
            bin = bin > NBINS - 1 ? NBINS - 1 : bin;
            atomicAdd(&hist[bin], 1u);
        }
    }
    __syncthreads();

    // Find cutoff bin so that count(bins > cut) < PRE <= count(bins >= cut).
    if (tid == 0) {
        unsigned int acc = 0; int cut = -1; int needEq = 0;
        for (int bb = NBINS - 1; bb >= 0; bb--) {
            unsigned int na = acc;
            acc += hist[bb];
            if (acc >= PRE) { cut = bb; needEq = PRE - (int)na; break; }
        }
        sCut = cut; sNeedEq = needEq; sRunA = 0; sRunE = 0;
    }
    __syncthreads();

    // Pass 2: deterministic collection in index order via block prefix scans.
    const int lane = tid & 31, wid = tid >> 5;
    const unsigned long long lmask = (1ull << lane) - 1ull;
    for (int base = 0; base < N; base += 1024) {
        int i = base + tid;
        float sc = 0.0f; int bin = -1;
        if (i < N) {
            sc = sigmoidf_(clsB[(size_t)i * NUM_CLASS + kcls]);
            if (sc >= SCORE_THRESH) {
                bin = (int)(sc * (float)NBINS);
                bin = bin > NBINS - 1 ? NBINS - 1 : bin;
            }
        }
        int flagA = (bin > sCut) ? 1 : 0;                 // strictly above cutoff: always taken
        int flagE = (bin >= 0 && bin == sCut) ? 1 : 0;    // cutoff bin: first sNeedEq in index order

        // block exclusive scan of flagA
        unsigned long long bal = __ballot(flagA);
        int pfx = __popcll(bal & lmask);
        if (lane == 0) wsum[wid] = (unsigned int)__popcll(bal);
        __syncthreads();
        if (tid == 0) { unsigned int a = 0; for (int w = 0; w < 32; w++) { unsigned int t = wsum[w]; wsum[w] = a; a += t; } wsum[32] = a; }
        __syncthreads();
        int posA = sRunA + (int)wsum[wid] + pfx;
        int totA = (int)wsum[32];
        if (flagA && posA < PRE) { sSc[posA] = sc; sIx[posA] = i; }
        __syncthreads();

        // block exclusive scan of flagE (cutoff-bin entries fill from the top end)
        bal = __ballot(flagE);
        pfx = __popcll(bal & lmask);
        if (lane == 0) wsum[wid] = (unsigned int)__popcll(bal);
        __syncthreads();
        if (tid == 0) { unsigned int a = 0; for (int w = 0; w < 32; w++) { unsigned int t = wsum[w]; wsum[w] = a; a += t; } wsum[32] = a; }
        __syncthreads();
        int eIdx = sRunE + (int)wsum[wid] + pfx;
        int totE = (int)wsum[32];
        if (flagE && eIdx < sNeedEq) { int sl = PRE - 1 - eIdx; sSc[sl] = sc; sIx[sl] = i; }
        __syncthreads();
        if (tid == 0) { sRunA += totA; sRunE += totE; }
        __syncthreads();
    }

    // Bitonic sort, descending by score (index-ascending tiebreak) — fixed network.
    for (int ksz = 2; ksz <= PRE; ksz <<= 1) {
        for (int j = ksz >> 1; j > 0; j >>= 1) {
            __syncthreads();
            if (tid < PRE) {
                int p = tid ^ j;
                if (p > tid) {
                    float s1 = sSc[tid], s2 = sSc[p];
                    int i1 = sIx[tid], i2 = sIx[p];
                    bool before = (s1 > s2) || (s1 == s2 && i1 < i2);
                    bool desc = (tid & ksz) == 0;
                    if (desc ? !before : before) {
                        sSc[tid] = s2; sSc[p] = s1; sIx[tid] = i2; sIx[p] = i1;
                    }
                }
            }
        }
    }
    __syncthreads();

    // Gather selected boxes into workspace (padding -> far-away dummy box so the
    // WMMA distance prefilter rejects every pair involving it).
    if (tid < PRE) {
        int gi = sIx[tid];
        int o = slot * PRE + tid;
        wsScore[o] = sSc[tid];
        float bx[7] = {1.0e8f, 1.0e8f, 0.0f, 1.0f, 1.0f, 1.0f, 0.0f};
        if (gi >= 0) {
            const float* src = box + ((size_t)b * N + gi) * 7;
#pragma unroll
            for (int c = 0; c < 7; c++) bx[c] = src[c];
        }
#pragma unroll
        for (int c = 0; c < 7; c++) wsBox[(size_t)o * 7 + c] = bx[c];
    }
}

// ---------------------------------------------------------------------------
// Kernel 2: per slot — WMMA center-distance Gram prefilter over the 512x512
// pair space (V_WMMA_F32_16X16X4_F32, one instruction per 16x16 tile; A-frag
// reused across a whole tile row), exact rotated 3D-IoU only for surviving
// pairs (~2%), LDS bit-matrix, serial greedy NMS scan, padded output assembly.
// One 256-thread workgroup (8 wave32) per slot.
// ---------------------------------------------------------------------------
__global__ __launch_bounds__(256) void nms_out_kernel(
    const float* __restrict__ wsScore, const float* __restrict__ wsBox,
    float* __restrict__ outBoxes, float* __restrict__ outScores,
    int* __restrict__ outLabels)
{
    __shared__ float sBox[PRE * 7];
    __shared__ float sN2[PRE], sRad[PRE], sSc[PRE];
    __shared__ unsigned int sMask[PRE * 16];   // 512 rows x 512 bits
    __shared__ int sKeep[PRE], sRank[PRE];
    __shared__ int sTot;

    const int tid = threadIdx.x;
    const int slot = blockIdx.x;
    const int b = slot / NUM_CLASS, kcls = slot % NUM_CLASS;

    for (int i = tid; i < PRE; i += 256) {
        int o = slot * PRE + i;
#pragma unroll
        for (int c = 0; c < 7; c++) sBox[i * 7 + c] = wsBox[(size_t)o * 7 + c];
        float x = sBox[i * 7 + 0], y = sBox[i * 7 + 1];
        float dx = sBox[i * 7 + 3], dy = sBox[i * 7 + 4];
        sN2[i] = x * x + y * y;
        sRad[i] = 0.5f * sqrtf(dx * dx + dy * dy);   // BEV half-diagonal
        sSc[i] = wsScore[o];
    }
    for (int i = tid; i < PRE * 16; i += 256) sMask[i] = 0;
    __syncthreads();

    const int lane = tid & 31;
    const int wv = tid >> 5;              // wave id 0..7 (uniform per wave)
    const bool hi = (lane & 16) != 0;     // lanes 16-31 carry K=2,3 (zero padding)
    const int l15 = lane & 15;
    const float zmul = hi ? 0.0f : 1.0f;

    // 32x32 tiles of the 512x512 pair space. Wave handles 4 tile-rows; the A
    // fragment is loaded once per row and reused across 32 WMMAs. Frag loads are
    // unconditional LDS reads (select-to-zero afterwards) so no EXEC juggling
    // sits on the WMMA path.
    for (int tm = wv; tm < 32; tm += 8) {
        int mi = tm * 16 + l15;
        v2f afrag;
        afrag[0] = sBox[mi * 7 + 0] * zmul;   // A: K0 = x_row (lanes 16-31 -> 0)
        afrag[1] = sBox[mi * 7 + 1] * zmul;   // A: K1 = y_row
        for (int tn = 0; tn < 32; ++tn) {
            int ni = tn * 16 + l15;
            v2f bfrag;
            bfrag[0] = sBox[ni * 7 + 0] * zmul;   // B: K0 row = x_col
            bfrag[1] = sBox[ni * 7 + 1] * zmul;   // B: K1 row = y_col
            v8f acc = {};
            acc = __builtin_amdgcn_wmma_f32_16x16x4_f32(
                false, afrag, false, bfrag, (short)0, acc, false, false);
            float garr[8];
#pragma unroll
            for (int e = 0; e < 8; e++) garr[e] = acc[e];
            int Nc = tn * 16 + l15;
            float n2n = sN2[Nc], radn = sRad[Nc];
            for (int e = 0; e < 8; e++) {       // not unrolled: one IoU call site
                int M = tm * 16 + e + (hi ? 8 : 0);
                float d2 = sN2[M] + n2n - 2.0f * garr[e];   // |c_M - c_N|^2
                float rr = sRad[M] + radn;
                if (M != Nc && d2 <= rr * rr + 1.0f) {      // slack: no false rejects
                    float iou = iou3d_pair(&sBox[M * 7], &sBox[Nc * 7]);
                    if (iou > NMS_THRESH)
                        atomicOr(&sMask[M * 16 + (Nc >> 5)], 1u << (Nc & 31));
                }
            }
        }
    }
    __syncthreads();

    // Greedy suppression scan (scores already sorted descending).
    if (tid == 0) {
        unsigned int sup[16];
#pragma unroll
        for (int w = 0; w < 16; w++) sup[w] = 0;
        int rank = 0;
        for (int i = 0; i < PRE; i++) {
            bool valid = sSc[i] >= SCORE_THRESH;
            bool kp = valid && !((sup[i >> 5] >> (i & 31)) & 1u);
            sKeep[i] = kp ? 1 : 0;
            sRank[i] = (kp && rank < POST) ? rank : -1;
            if (kp) {
                rank++;
#pragma unroll
                for (int w = 0; w < 16; w++) sup[w] |= sMask[i * 16 + w];
            }
        }
        sTot = rank < POST ? rank : POST;
    }
    __syncthreads();

    // Write outputs: kept rows to slots [0, sTot), zeros to [sTot, POST) — disjoint.
    const int obase = b * (NUM_CLASS * POST) + kcls * POST;
    for (int i = tid; i < PRE; i += 256) {
        int r = sRank[i];
        if (sKeep[i] && r >= 0) {
            int o = obase + r;
#pragma unroll
            for (int c = 0; c < 7; c++) outBoxes[(size_t)o * 7 + c] = sBox[i * 7 + c];
            outScores[o] = sSc[i];
            outLabels[o] = kcls + 1;
        }
    }
    for (int j = tid; j < POST; j += 256) {
        if (j >= sTot) {
            int o = obase + j;
#pragma unroll
            for (int c = 0; c < 7; c++) outBoxes[(size_t)o * 7 + c] = 0.0f;
            outScores[o] = 0.0f;
            outLabels[o] = 0;
        }
    }
}

extern "C" void kernel_launch(void* const* d_in, const int* in_sizes, int n_in,
                              void* d_out, int out_size, void* d_ws, size_t ws_size,
                              hipStream_t stream) {
    (void)n_in; (void)out_size; (void)ws_size;
    const float* cls = (const float*)d_in[0];   // (B, N, 3) f32
    const float* box = (const float*)d_in[1];   // (B, N, 7) f32
    const int B = 2;
    const int N = in_sizes[0] / (B * NUM_CLASS);

    // Workspace: 6*512 scores (12 KB) + 6*512*7 box floats (84 KB) = 96 KB.
    float* wsScore = (float*)d_ws;
    float* wsBox = wsScore + (size_t)B * NUM_CLASS * PRE;

    float* out = (float*)d_out;
    float* outBoxes = out;                                          // 2*300*7
    float* outScores = out + (size_t)B * NUM_CLASS * POST * 7;      // 2*300
    int* outLabels = (int*)(out + (size_t)B * NUM_CLASS * POST * 7
                                + (size_t)B * NUM_CLASS * POST);    // 2*300 (int32)

    topk_kernel<<<B * NUM_CLASS, 1024, 0, stream>>>(cls, box, wsScore, wsBox, N);
    nms_out_kernel<<<B * NUM_CLASS, 256, 0, stream>>>(wsScore, wsBox,
                                                      outBoxes, outScores, outLabels);
}